// GCN_61967788147082
// MI455X (gfx1250) — compile-verified
//
#include <hip/hip_runtime.h>
#include <hip/hip_bf16.h>

#define N_NODES 20000
#define N_EDGES 320000
#define F_IN    300
#define DHID    1024
#define N_CLS   10

typedef float v2f __attribute__((ext_vector_type(2)));
typedef float v8f __attribute__((ext_vector_type(8)));

// ---------------------------------------------------------------------------
// CSR construction: counts -> exclusive scan -> cursor fill
// ---------------------------------------------------------------------------
__global__ void zero_ints(int* __restrict__ p, int n) {
    int i = blockIdx.x * blockDim.x + threadIdx.x;
    if (i < n) p[i] = 0;
}

__global__ void hist_dst(const int* __restrict__ dst, int* __restrict__ cnt, int e) {
    int i = blockIdx.x * blockDim.x + threadIdx.x;
    if (i < e) atomicAdd(&cnt[dst[i]], 1);
}

// Single-block exclusive scan over N_NODES counts -> offs[0..N_NODES]
#define SCAN_T 256
#define SCAN_C 80   // 256*80 = 20480 >= N_NODES
__global__ void scan_offsets(const int* __restrict__ cnt, int* __restrict__ offs) {
    __shared__ int part[SCAN_T];
    int t = threadIdx.x;
    int base = t * SCAN_C;
    int s = 0;
    for (int i = 0; i < SCAN_C; ++i) {
        int idx = base + i;
        s += (idx < N_NODES) ? cnt[idx] : 0;
    }
    part[t] = s;
    __syncthreads();
    for (int off = 1; off < SCAN_T; off <<= 1) {   // Hillis-Steele inclusive scan
        int v = (t >= off) ? part[t - off] : 0;
        __syncthreads();
        part[t] += v;
        __syncthreads();
    }
    int run = (t == 0) ? 0 : part[t - 1];
    for (int i = 0; i < SCAN_C; ++i) {
        int idx = base + i;
        if (idx < N_NODES) { offs[idx] = run; run += cnt[idx]; }
    }
    if (t == SCAN_T - 1) offs[N_NODES] = run;
}

// dis[i] = (1 + indeg)^{-1/2}
__global__ void make_dis(const int* __restrict__ cnt, float* __restrict__ dis, int n) {
    int i = blockIdx.x * blockDim.x + threadIdx.x;
    if (i < n) dis[i] = rsqrtf(1.0f + (float)cnt[i]);
}

__global__ void fill_csr(const int* __restrict__ src, const int* __restrict__ dst,
                         const int* __restrict__ offs, int* __restrict__ cursor,
                         int* __restrict__ csr_src, int e) {
    int i = blockIdx.x * blockDim.x + threadIdx.x;
    if (i >= e) return;
    int d = dst[i];
    int pos = atomicAdd(&cursor[d], 1);
    csr_src[offs[d] + pos] = src[i];
}

// ---------------------------------------------------------------------------
// f32 WMMA GEMM: C[M,N] = A[M,K] * B[K,N]
// One wave computes a 32x64 tile: 2 (M) x 4 (N) accumulators of 16x16, K-step 4.
// Each B fetch feeds two WMMAs (A row-pair reuse): ~10.7 flop/byte from cache.
// VGPR layouts per CDNA5 ISA 7.12.2:
//   A 16x4 : lane<16 -> {K=0,K=1}, lane>=16 -> {K=2,K=3}, M = lane&15
//   B 4x16 : VGPR0 rows {0 | 2}, VGPR1 rows {1 | 3} by lane half, N = lane&15
//   C 16x16: VGPR r -> row r (lanes 0-15) / row r+8 (lanes 16-31)
// ---------------------------------------------------------------------------
__global__ void gemm_wmma_f32(const float* __restrict__ A,
                              const float* __restrict__ B,
                              float* __restrict__ C,
                              int M, int N, int K) {
    const int wavesPerBlk = blockDim.x >> 5;
    const int wave  = blockIdx.x * wavesPerBlk + (threadIdx.x >> 5);
    const int tilesN = N >> 6;                       // tiles of 64 cols
    const int totalWaves = (M >> 5) * tilesN;        // 32 rows per wave
    if (wave >= totalWaves) return;

    const int tm = (wave / tilesN) << 5;
    const int tn = (wave % tilesN) << 6;

    const int lane = threadIdx.x & 31;
    const int half = lane >> 4;
    const int lo   = lane & 15;

    v8f c00 = {}, c01 = {}, c02 = {}, c03 = {};
    v8f c10 = {}, c11 = {}, c12 = {}, c13 = {};

    const float* arow0 = A + (size_t)(tm + lo) * K + 2 * half;   // 8B aligned (K even)
    const float* arow1 = arow0 + (size_t)16 * K;
    const float* bcol  = B + (size_t)(2 * half) * N + tn + lo;

    for (int k = 0; k < K; k += 4) {
        v2f a0 = *(const v2f*)(arow0 + k);
        v2f a1 = *(const v2f*)(arow1 + k);

        const float* bk = bcol + (size_t)k * N;
        v2f b0, b1, b2, b3;
        b0.x = bk[0];       b0.y = bk[N];
        b1.x = bk[16];      b1.y = bk[N + 16];
        b2.x = bk[32];      b2.y = bk[N + 32];
        b3.x = bk[48];      b3.y = bk[N + 48];

        c00 = __builtin_amdgcn_wmma_f32_16x16x4_f32(false, a0, false, b0, (short)0, c00, false, false);
        c10 = __builtin_amdgcn_wmma_f32_16x16x4_f32(false, a1, false, b0, (short)0, c10, false, false);
        c01 = __builtin_amdgcn_wmma_f32_16x16x4_f32(false, a0, false, b1, (short)0, c01, false, false);
        c11 = __builtin_amdgcn_wmma_f32_16x16x4_f32(false, a1, false, b1, (short)0, c11, false, false);
        c02 = __builtin_amdgcn_wmma_f32_16x16x4_f32(false, a0, false, b2, (short)0, c02, false, false);
        c12 = __builtin_amdgcn_wmma_f32_16x16x4_f32(false, a1, false, b2, (short)0, c12, false, false);
        c03 = __builtin_amdgcn_wmma_f32_16x16x4_f32(false, a0, false, b3, (short)0, c03, false, false);
        c13 = __builtin_amdgcn_wmma_f32_16x16x4_f32(false, a1, false, b3, (short)0, c13, false, false);
    }

    float* crow0 = C + (size_t)(tm + 8 * half) * N + tn + lo;
    float* crow1 = crow0 + (size_t)16 * N;
#pragma unroll
    for (int r = 0; r < 8; ++r) {
        float* c0 = crow0 + (size_t)r * N;
        float* c1 = crow1 + (size_t)r * N;
        c0[0]  = c00[r];  c0[16] = c01[r];  c0[32] = c02[r];  c0[48] = c03[r];
        c1[0]  = c10[r];  c1[16] = c11[r];  c1[32] = c12[r];  c1[48] = c13[r];
    }
}

// ---------------------------------------------------------------------------
// Fused GCN aggregation per node (one block = one node, 256 thr x float4):
//   z[i] = relu( sum_{e: dst=i} h[src_e]*dis[src_e]*dis[i]  +  h[i]*dis[i]^2 + b )
// All rows live in L2 (H is 82MB vs 192MB L2): pure coalesced gather, no atomics.
// ---------------------------------------------------------------------------
__global__ void gather_agg_relu(const float* __restrict__ H,
                                const int* __restrict__ csr_src,
                                const int* __restrict__ offs,
                                const float* __restrict__ dis,
                                const float* __restrict__ bias,
                                float* __restrict__ Z) {
    const int node = blockIdx.x;
    const int c4 = threadIdx.x << 2;         // 256 threads * 4 = 1024 cols
    const float di = dis[node];

    const int beg = offs[node], end = offs[node + 1];
    float4 acc = make_float4(0.f, 0.f, 0.f, 0.f);
    for (int i = beg; i < end; ++i) {
        int s = csr_src[i];
        float nrm = dis[s] * di;
        const float4 h = *(const float4*)(H + (size_t)s * DHID + c4);
        acc.x += h.x * nrm; acc.y += h.y * nrm;
        acc.z += h.z * nrm; acc.w += h.w * nrm;
    }
    const float sl = di * di;
    const float4 hs = *(const float4*)(H + (size_t)node * DHID + c4);
    const float4 b  = *(const float4*)(bias + c4);
    float4 o;
    o.x = fmaxf(acc.x + hs.x * sl + b.x, 0.f);
    o.y = fmaxf(acc.y + hs.y * sl + b.y, 0.f);
    o.z = fmaxf(acc.z + hs.z * sl + b.z, 0.f);
    o.w = fmaxf(acc.w + hs.w * sl + b.w, 0.f);
    *(float4*)(Z + (size_t)node * DHID + c4) = o;
}

// ---------------------------------------------------------------------------
// logits = Z @ Wfc + bfc ; out = softmax(logits).  One wave per node.
// ---------------------------------------------------------------------------
__global__ void fc_softmax(const float* __restrict__ Z, const float* __restrict__ W,
                           const float* __restrict__ b, float* __restrict__ out) {
    int node = blockIdx.x * (blockDim.x >> 5) + (threadIdx.x >> 5);
    if (node >= N_NODES) return;
    int lane = threadIdx.x & 31;

    float acc[N_CLS];
#pragma unroll
    for (int c = 0; c < N_CLS; ++c) acc[c] = 0.f;

    const float* z = Z + (size_t)node * DHID;
    for (int k = lane; k < DHID; k += 32) {
        float zv = z[k];
        const float* w = W + (size_t)k * N_CLS;
#pragma unroll
        for (int c = 0; c < N_CLS; ++c) acc[c] += zv * w[c];
    }
#pragma unroll
    for (int c = 0; c < N_CLS; ++c)
        for (int off = 16; off > 0; off >>= 1)
            acc[c] += __shfl_down(acc[c], off, 32);

    if (lane == 0) {
        float l[N_CLS], m = -1e30f;
#pragma unroll
        for (int c = 0; c < N_CLS; ++c) { l[c] = acc[c] + b[c]; m = fmaxf(m, l[c]); }
        float s = 0.f;
#pragma unroll
        for (int c = 0; c < N_CLS; ++c) { l[c] = __expf(l[c] - m); s += l[c]; }
        float inv = 1.0f / s;
        float* o = out + (size_t)node * N_CLS;
#pragma unroll
        for (int c = 0; c < N_CLS; ++c) o[c] = l[c] * inv;
    }
}

// ---------------------------------------------------------------------------
// Host-side launch sequencing (all on `stream`, graph-capture safe)
// ---------------------------------------------------------------------------
extern "C" void kernel_launch(void* const* d_in, const int* in_sizes, int n_in,
                              void* d_out, int out_size, void* d_ws, size_t ws_size,
                              hipStream_t stream) {
    const float* x    = (const float*)d_in[0];
    const int*   ei   = (const int*)  d_in[1];   // [2, E]: src row then dst row
    const float* W1   = (const float*)d_in[2];
    const float* b1   = (const float*)d_in[3];
    const float* W2   = (const float*)d_in[4];
    const float* b2   = (const float*)d_in[5];
    const float* Wfc  = (const float*)d_in[6];
    const float* bfc  = (const float*)d_in[7];
    float* out = (float*)d_out;

    const int* src = ei;
    const int* dst = ei + N_EDGES;

    // Workspace layout (bytes):
    //   counts   @ 0         (20480 ints)
    //   cursors  @ 81920     (20480 ints)
    //   offs     @ 163840    (20001 ints)
    //   dis      @ 245760    (20000 f32)
    //   csr_src  @ 327680    (320000 ints)
    //   bufH     @ 2 MiB     (20000x1024 f32, 81.92 MB)
    //   bufZ     @ 2MiB+H    (20000x1024 f32)
    char* ws = (char*)d_ws;
    int*   counts  = (int*)(ws);
    int*   cursors = (int*)(ws + 81920);
    int*   offs    = (int*)(ws + 163840);
    float* dis     = (float*)(ws + 245760);
    int*   csr_src = (int*)(ws + 327680);
    float* bufH    = (float*)(ws + (2u << 20));
    float* bufZ    = bufH + (size_t)N_NODES * DHID;

    const int TPB = 256;
    const int nodeBlks = (N_NODES + TPB - 1) / TPB;
    const int edgeBlks = (N_EDGES + TPB - 1) / TPB;

    // CSR build + symmetric normalization
    zero_ints<<<nodeBlks, TPB, 0, stream>>>(counts,  N_NODES);
    zero_ints<<<nodeBlks, TPB, 0, stream>>>(cursors, N_NODES);
    hist_dst <<<edgeBlks, TPB, 0, stream>>>(dst, counts, N_EDGES);
    scan_offsets<<<1, SCAN_T, 0, stream>>>(counts, offs);
    make_dis <<<nodeBlks, TPB, 0, stream>>>(counts, dis, N_NODES);
    fill_csr <<<edgeBlks, TPB, 0, stream>>>(src, dst, offs, cursors, csr_src, N_EDGES);

    // ---- Layer 1 ----
    {
        int waves = (N_NODES / 32) * (DHID / 64);
        gemm_wmma_f32<<<waves / 8, TPB, 0, stream>>>(x, W1, bufH, N_NODES, DHID, F_IN);
    }
    gather_agg_relu<<<N_NODES, TPB, 0, stream>>>(bufH, csr_src, offs, dis, b1, bufZ);

    // ---- Layer 2 ----
    {
        int waves = (N_NODES / 32) * (DHID / 64);
        gemm_wmma_f32<<<waves / 8, TPB, 0, stream>>>(bufZ, W2, bufH, N_NODES, DHID, DHID);
    }
    gather_agg_relu<<<N_NODES, TPB, 0, stream>>>(bufH, csr_src, offs, dis, b2, bufZ);

    // ---- FC + softmax ----
    fc_softmax<<<(N_NODES + 7) / 8, TPB, 0, stream>>>(bufZ, Wfc, bfc, out);
}